// SparseBSRLinear_60687887893041
// MI455X (gfx1250) — compile-verified
//
#include <hip/hip_runtime.h>

typedef __attribute__((ext_vector_type(2))) float v2f;
typedef __attribute__((ext_vector_type(8))) float v8f;

#define N_TOKENS 8192
#define IN_F     4096
#define OUT_F    4096
#define BS       64
#define NB_OUT   (OUT_F / BS)   // 64 block rows
#define NB_IN    (IN_F / BS)    // 64 block cols
#define TM       128            // token tile per workgroup
#define LDT      (BS + 4)       // 68-float LDS row stride -> conflict-free fragment reads

// ---------------------------------------------------------------------------
// Kernel 1: build BSR index lists from the block mask (deterministic, no
// atomics: each of the 64 threads scans its block-row in order).
// ---------------------------------------------------------------------------
__global__ void bsr_build_index(const float* __restrict__ mask,
                                int* __restrict__ counts,
                                int* __restrict__ idx) {
  const int i = threadIdx.x;
  if (i >= NB_OUT) return;
  int c = 0;
  const size_t rowbase = (size_t)i * BS * IN_F;
  for (int j = 0; j < NB_IN; ++j) {
    if (mask[rowbase + (size_t)j * BS] != 0.0f) idx[i * NB_IN + c++] = j;
  }
  counts[i] = c;
}

// ---------------------------------------------------------------------------
// Kernel 2: block-sparse GEMM on the f32 WMMA pipe.
//   grid = (N_TOKENS/TM, NB_OUT), block = 256 threads (8 wave32s).
//   Workgroup tile: 128 tokens x 64 out-features. Wave tile: 16 x 64.
// ---------------------------------------------------------------------------
__global__ __launch_bounds__(256) void bsr_wmma_f32(
    const float* __restrict__ X, const float* __restrict__ W,
    const float* __restrict__ bias, const int* __restrict__ counts,
    const int* __restrict__ idx, float* __restrict__ out) {
  __shared__ float lX[TM * LDT];  // 128*68*4 = 34816 B
  __shared__ float lW[BS * LDT];  //  64*68*4 = 17408 B

  const int tid  = threadIdx.x;
  const int lane = tid & 31;      // wave32
  const int wave = tid >> 5;
  const int lm   = lane & 15;
  const int half = lane >> 4;     // ISA: lanes 16-31 carry K+2,K+3 (A/B frags)

  const int m0   = blockIdx.x * TM;
  const int brow = blockIdx.y;
  const int n0   = brow * BS;
  const int nk   = counts[brow];  // uniform across workgroup -> EXEC all-1 at WMMA
  const int* __restrict__ blist = idx + brow * NB_IN;

  v8f acc0 = {}, acc1 = {}, acc2 = {}, acc3 = {};

  for (int j = 0; j < nk; ++j) {
    const int k0 = blist[j] * BS;

    // Stage X tile [TM x 64] and W tile [64 x 64] into LDS, coalesced float4.
    for (int i = tid; i < TM * (BS / 4); i += 256) {
      const int r = i >> 4, c = (i & 15) << 2;
      *(float4*)(&lX[r * LDT + c]) =
          *(const float4*)(X + (size_t)(m0 + r) * IN_F + k0 + c);
    }
    for (int i = tid; i < BS * (BS / 4); i += 256) {
      const int r = i >> 4, c = (i & 15) << 2;
      *(float4*)(&lW[r * LDT + c]) =
          *(const float4*)(W + (size_t)(n0 + r) * IN_F + k0 + c);
    }

    // Prefetch the next active block into cache while this one is consumed.
    if (j + 1 < nk) {
      const int kn = blist[j + 1] * BS;
      __builtin_prefetch(X + (size_t)(m0 + (tid >> 1)) * IN_F + kn + (tid & 1) * 32, 0, 0);
      if (tid < 2 * BS)
        __builtin_prefetch(W + (size_t)(n0 + (tid >> 1)) * IN_F + kn + (tid & 1) * 32, 0, 0);
    }
    __syncthreads();

    // Fragment base pointers. A: row (wave*16+lm) of X tile. B: row lm of the
    // W tile (B = W^T, so B[k][n] = W[n][k], contiguous in k).
    const float* xa = &lX[(wave * 16 + lm) * LDT + 2 * half];
    const float* wb = &lW[lm * LDT + 2 * half];

#pragma unroll
    for (int kk = 0; kk < BS; kk += 4) {
      const v2f a  = *(const v2f*)(xa + kk);
      const v2f b0 = *(const v2f*)(wb + kk);
      const v2f b1 = *(const v2f*)(wb + 16 * LDT + kk);
      const v2f b2 = *(const v2f*)(wb + 32 * LDT + kk);
      const v2f b3 = *(const v2f*)(wb + 48 * LDT + kk);
      acc0 = __builtin_amdgcn_wmma_f32_16x16x4_f32(false, a, false, b0, (short)0, acc0, false, false);
      acc1 = __builtin_amdgcn_wmma_f32_16x16x4_f32(false, a, false, b1, (short)0, acc1, false, false);
      acc2 = __builtin_amdgcn_wmma_f32_16x16x4_f32(false, a, false, b2, (short)0, acc2, false, false);
      acc3 = __builtin_amdgcn_wmma_f32_16x16x4_f32(false, a, false, b3, (short)0, acc3, false, false);
    }
    __syncthreads();
  }

  // Epilogue: C/D layout is VGPR r -> row (r + 8*half), col = lm within tile.
  const int rowbase = m0 + wave * 16 + half * 8;
  v8f accs[4] = {acc0, acc1, acc2, acc3};
#pragma unroll
  for (int t = 0; t < 4; ++t) {
    const int col = n0 + t * 16 + lm;
    const float bv = bias[col];
#pragma unroll
    for (int r = 0; r < 8; ++r)
      out[(size_t)(rowbase + r) * OUT_F + col] = accs[t][r] + bv;
  }
}

extern "C" void kernel_launch(void* const* d_in, const int* in_sizes, int n_in,
                              void* d_out, int out_size, void* d_ws, size_t ws_size,
                              hipStream_t stream) {
  (void)in_sizes; (void)n_in; (void)out_size; (void)ws_size;
  const float* X    = (const float*)d_in[0];
  const float* W    = (const float*)d_in[1];
  const float* bias = (const float*)d_in[2];
  const float* mask = (const float*)d_in[3];
  float* out = (float*)d_out;

  int* counts = (int*)d_ws;            // 64 ints
  int* idx    = counts + NB_OUT;       // 64 x 64 ints

  bsr_build_index<<<1, NB_OUT, 0, stream>>>(mask, counts, idx);
  dim3 grid(N_TOKENS / TM, NB_OUT);
  bsr_wmma_f32<<<grid, 256, 0, stream>>>(X, W, bias, counts, idx, out);
}